// myGATConv_73684458930528
// MI455X (gfx1250) — compile-verified
//
#include <hip/hip_runtime.h>
#include <hip/hip_bf16.h>

typedef __attribute__((ext_vector_type(2))) float v2f;
typedef __attribute__((ext_vector_type(8))) float v8f;

#define IN_F      256
#define OUT_TOT   256   // NHEAD * OUT_F
#define NHEAD     8
#define DHEAD     32
#define EDGE_F    32
#define NEG_SLOPE 0.1f

// ---------------- W transpose: Wt[n][k] = W[k][n], 256x256 ----------------
__global__ void transposeW(const float* __restrict__ W, float* __restrict__ Wt) {
    int idx = blockIdx.x * 256 + threadIdx.x;   // 65536 threads
    int k = idx >> 8, n = idx & 255;
    Wt[n * 256 + k] = W[k * 256 + n];
}

// ---------------- h = nan_guard(x @ W) via V_WMMA_F32_16X16X4_F32 ----------
// One wave (32 lanes) computes a 16-row x 256-col block of h.
// A layout (f32 16x4): lanes 0-15 -> K pair {0,1}, lanes 16-31 -> K pair {2,3}.
// B layout mirrored (from transposed W so each lane's K-pair is contiguous).
__global__ void __launch_bounds__(32)
gemm_h(const float* __restrict__ x, const float* __restrict__ Wt,
       float* __restrict__ h, int n_nodes) {
    const int lane = threadIdx.x;
    const int m0   = blockIdx.x * 16;
    const int half = lane >> 4;        // 0 -> K{0,1}, 1 -> K{2,3}
    const int l15  = lane & 15;

    int mrow = m0 + l15;
    if (mrow >= n_nodes) mrow = n_nodes - 1;     // clamp loads, keep EXEC full
    const float* xrow = x + (long)mrow * IN_F;

    const v8f zero = {0.f, 0.f, 0.f, 0.f, 0.f, 0.f, 0.f, 0.f};
    v8f acc[16];
#pragma unroll
    for (int t = 0; t < 16; ++t) acc[t] = zero;

    for (int k = 0; k < IN_F; k += 4) {
        v2f a = *(const v2f*)(xrow + k + 2 * half);
#pragma unroll
        for (int t = 0; t < 16; ++t) {
            v2f b = *(const v2f*)(Wt + (long)(t * 16 + l15) * 256 + k + 2 * half);
            acc[t] = __builtin_amdgcn_wmma_f32_16x16x4_f32(
                false, a, false, b, (short)0, acc[t], false, false);
        }
    }

    // C/D layout: VGPR v -> row m0 + v + 8*half, col t*16 + l15
#pragma unroll
    for (int t = 0; t < 16; ++t) {
#pragma unroll
        for (int v = 0; v < 8; ++v) {
            int r = m0 + v + 8 * half;
            if (r < n_nodes) {
                float val = acc[t][v];
                val = (val != val) ? 0.0f : val;     // NaN guard
                h[(long)r * OUT_TOT + t * 16 + l15] = val;
            }
        }
    }
}

// ---------------- hl/hr: per (node, head) attention logits ----------------
__global__ void node_logits(const float* __restrict__ h,
                            const float* __restrict__ a_l,
                            const float* __restrict__ a_r,
                            float* __restrict__ hl, float* __restrict__ hr,
                            int n_nodes) {
    long idx = (long)blockIdx.x * blockDim.x + threadIdx.x;
    if (idx >= (long)n_nodes * NHEAD) return;
    int n = (int)(idx >> 3), hd = (int)(idx & 7);
    const float* hp = h + (long)n * OUT_TOT + hd * DHEAD;
    const float* al = a_l + hd * DHEAD;
    const float* ar = a_r + hd * DHEAD;
    float sl = 0.f, sr = 0.f;
#pragma unroll
    for (int d = 0; d < DHEAD; ++d) {
        float v = hp[d];
        sl += al[d] * v;
        sr += ar[d] * v;
    }
    hl[idx] = sl;
    hr[idx] = sr;
}

// ---------------- he[t][hd] = sum_fe a_e * (edge_emb @ W_e) ---------------
__global__ void edge_he(const float* __restrict__ edge_emb,
                        const float* __restrict__ W_e,
                        const float* __restrict__ a_e,
                        float* __restrict__ he, int n_et) {
    int t = threadIdx.x >> 3, hd = threadIdx.x & 7;
    if (t >= n_et) return;
    float s = 0.f;
    for (int fe = 0; fe < EDGE_F; ++fe) {
        float acc = 0.f;
        for (int k = 0; k < EDGE_F; ++k)
            acc += edge_emb[t * EDGE_F + k] * W_e[k * (NHEAD * EDGE_F) + hd * EDGE_F + fe];
        s += a_e[hd * EDGE_F + fe] * acc;
    }
    he[t * NHEAD + hd] = s;
}

// ---------------- init: m = -inf, z = 0, out = 0 --------------------------
__global__ void init_bufs(float* __restrict__ m, float* __restrict__ z,
                          float* __restrict__ out, long nh, long outsz) {
    long idx = (long)blockIdx.x * blockDim.x + threadIdx.x;
    if (idx < outsz) out[idx] = 0.f;
    if (idx < nh) { m[idx] = -__builtin_inff(); z[idx] = 0.f; }
}

// ---------------- mixed-sign float atomic max via int/uint trick ----------
__device__ __forceinline__ void atomicMaxF(float* addr, float val) {
    if (val >= 0.f) atomicMax((int*)addr, __float_as_int(val));
    else            atomicMin((unsigned int*)addr, __float_as_uint(val));
}

// ---------------- s = leaky(hl[row]+hr[col]+he[tp]); segment max ----------
__global__ void edge_scores(const int* __restrict__ row, const int* __restrict__ col,
                            const int* __restrict__ tp,
                            const float* __restrict__ hl, const float* __restrict__ hr,
                            const float* __restrict__ he,
                            float* __restrict__ sbuf, float* __restrict__ m, long nE) {
    long idx = (long)blockIdx.x * blockDim.x + threadIdx.x;
    if (idx >= nE * NHEAD) return;
    long e = idx >> 3;
    int hd = (int)(idx & 7);
    float s = hl[(long)row[e] * NHEAD + hd] +
              hr[(long)col[e] * NHEAD + hd] +
              he[tp[e] * NHEAD + hd];
    s = s > 0.f ? s : NEG_SLOPE * s;
    sbuf[idx] = s;
    atomicMaxF(m + (long)col[e] * NHEAD + hd, s);
}

// ---------------- ex = exp(s - m[col]); z = segment sum -------------------
__global__ void edge_exp(const int* __restrict__ col, const float* __restrict__ sbuf,
                         const float* __restrict__ m, float* __restrict__ att,
                         float* __restrict__ z, long nE) {
    long idx = (long)blockIdx.x * blockDim.x + threadIdx.x;
    if (idx >= nE * NHEAD) return;
    long e = idx >> 3;
    int hd = (int)(idx & 7);
    float ex = __expf(sbuf[idx] - m[(long)col[e] * NHEAD + hd]);
    att[idx] = ex;
    atomicAdd(z + (long)col[e] * NHEAD + hd, ex);
}

// ---------------- normalize att in-place + scatter out --------------------
// One wave per edge; lane l covers d = l + 32*j (head j), giving 8 coalesced
// 128B gather reads of h[row] and 8 coalesced f32 atomic-add bursts to out[col].
__global__ void edge_aggregate(const int* __restrict__ row, const int* __restrict__ col,
                               const float* __restrict__ h, const float* __restrict__ zbuf,
                               float* __restrict__ att, float* __restrict__ out, long nE) {
    int lane = threadIdx.x & 31;
    long e = ((long)blockIdx.x * blockDim.x + threadIdx.x) >> 5;
    if (e >= nE) return;
    int r = row[e], c = col[e];
    float an[NHEAD];
#pragma unroll
    for (int j = 0; j < NHEAD; ++j)
        an[j] = att[e * NHEAD + j] / zbuf[(long)c * NHEAD + j];
    if (lane < NHEAD) att[e * NHEAD + lane] = an[lane];   // write normalized att
    const float* hp = h + (long)r * OUT_TOT;
    float* op = out + (long)c * OUT_TOT;
#pragma unroll
    for (int j = 0; j < NHEAD; ++j)
        atomicAdd(op + lane + 32 * j, an[j] * hp[lane + 32 * j]);
}

// ==========================================================================
extern "C" void kernel_launch(void* const* d_in, const int* in_sizes, int n_in,
                              void* d_out, int out_size, void* d_ws, size_t ws_size,
                              hipStream_t stream) {
    const float* x        = (const float*)d_in[0];
    const float* W        = (const float*)d_in[1];
    const float* W_e      = (const float*)d_in[2];
    const float* edge_emb = (const float*)d_in[3];
    const float* a_l      = (const float*)d_in[4];
    const float* a_r      = (const float*)d_in[5];
    const float* a_e      = (const float*)d_in[6];
    const int*   row      = (const int*)d_in[7];
    const int*   col      = (const int*)d_in[8];
    const int*   tp       = (const int*)d_in[9];

    const int  n_nodes = in_sizes[0] / IN_F;
    const long nE      = in_sizes[7];
    const int  n_et    = in_sizes[3] / EDGE_F;

    // workspace layout (floats)
    float* ws   = (float*)d_ws;
    float* h    = ws;                              // n_nodes * 256
    float* Wt   = h    + (long)n_nodes * OUT_TOT;  // 65536
    float* hl   = Wt   + IN_F * OUT_TOT;           // n_nodes * 8
    float* hr   = hl   + (long)n_nodes * NHEAD;
    float* he   = hr   + (long)n_nodes * NHEAD;    // n_et * 8
    float* mbuf = he   + (long)n_et * NHEAD;       // n_nodes * 8
    float* zbuf = mbuf + (long)n_nodes * NHEAD;
    float* sbuf = zbuf + (long)n_nodes * NHEAD;    // nE * 8

    float* out = (float*)d_out;                    // n_nodes * 256
    float* att = out + (long)n_nodes * OUT_TOT;    // nE * 8

    const long outsz = (long)n_nodes * OUT_TOT;
    const long nh    = (long)n_nodes * NHEAD;

    transposeW<<<IN_F, 256, 0, stream>>>(W, Wt);
    gemm_h<<<(n_nodes + 15) / 16, 32, 0, stream>>>(x, Wt, h, n_nodes);
    node_logits<<<(unsigned)((nh + 255) / 256), 256, 0, stream>>>(h, a_l, a_r, hl, hr, n_nodes);
    edge_he<<<1, 64, 0, stream>>>(edge_emb, W_e, a_e, he, n_et);
    init_bufs<<<(unsigned)((outsz + 255) / 256), 256, 0, stream>>>(mbuf, zbuf, out, nh, outsz);
    edge_scores<<<(unsigned)((nE * NHEAD + 255) / 256), 256, 0, stream>>>(row, col, tp, hl, hr, he, sbuf, mbuf, nE);
    edge_exp<<<(unsigned)((nE * NHEAD + 255) / 256), 256, 0, stream>>>(col, sbuf, mbuf, att, zbuf, nE);
    edge_aggregate<<<(unsigned)((nE * 32 + 255) / 256), 256, 0, stream>>>(row, col, h, zbuf, att, out, nE);
}